// vMF_6622839571153
// MI455X (gfx1250) — compile-verified
//
#include <hip/hip_runtime.h>
#include <math.h>

typedef __attribute__((ext_vector_type(2))) float v2f;
typedef __attribute__((ext_vector_type(8))) float v8f;

#define EMBED      512
#define PAD_INDEX  1
#define KCHUNKS    (EMBED / 4)   // 128 chunks of K=4
#define KGROUPS    (KCHUNKS / 8) // 16 groups; each group = one 128B line per row-half
#define WAVES_PER_BLOCK 8

// One wave32 processes 16 tokens. For each K-chunk of 4:
//   A (16x4 f32): lane l holds M = l%16, VGPR v holds K = v + 2*(l/16)  [ISA 7.12.2]
//   B (4x16 f32): lane l holds N = l%16, VGPR v holds K = v + 2*(l/16)  [mirrors documented 16-bit B]
// => identical register image; one float2 per lane serves as both operands.
__global__ __launch_bounds__(256) void vmf_tile_kernel(
    const float* __restrict__ outputs,      // [NTOK, 512]
    const int*   __restrict__ targets,      // [NTOK]
    const float* __restrict__ table,        // [32000, 512]
    float*       __restrict__ partial,      // [ntiles]
    int ntiles)
{
    const int lane = threadIdx.x & 31;
    const int wave = threadIdx.x >> 5;
    const int tile = blockIdx.x * WAVES_PER_BLOCK + wave;
    if (tile >= ntiles) return;             // wave-uniform: EXEC stays all-1s inside

    const int tokBase = tile * 16;
    const int m = lane & 15;                // token within tile (M for A, N for B)
    const int h = lane >> 4;                // lane-half -> K offset 0 or 2

    const int    tok  = tokBase + m;
    const float* orow = outputs + (size_t)tok * EMBED + 2 * h;
    const int    tgt  = targets[tok];
    const float* trow = table + (size_t)tgt * EMBED + 2 * h;

    v8f acc_oo = {0.f, 0.f, 0.f, 0.f, 0.f, 0.f, 0.f, 0.f};
    v8f acc_tt = {0.f, 0.f, 0.f, 0.f, 0.f, 0.f, 0.f, 0.f};
    v8f acc_ot = {0.f, 0.f, 0.f, 0.f, 0.f, 0.f, 0.f, 0.f};

    for (int g = 0; g < KGROUPS; ++g) {
        const float* ob = orow + 32 * g;    // this group's 128B line (per lane walk)
        const float* tb = trow + 32 * g;
        __builtin_prefetch(ob + 32, 0, 3);  // next line, near (WGP) scope
        __builtin_prefetch(tb + 32, 0, 3);
        #pragma unroll
        for (int u = 0; u < 8; ++u) {
            v2f a = *(const v2f*)(ob + 4 * u);   // {K = k0+2h, K = k0+2h+1}
            v2f b = *(const v2f*)(tb + 4 * u);
            // D = A*B + C ; diag(o*o)=|out|^2, diag(t*t)=|tar|^2, diag(o*t)=<out,tar>
            acc_oo = __builtin_amdgcn_wmma_f32_16x16x4_f32(false, a, false, a, (short)0, acc_oo, false, false);
            acc_tt = __builtin_amdgcn_wmma_f32_16x16x4_f32(false, b, false, b, (short)0, acc_tt, false, false);
            acc_ot = __builtin_amdgcn_wmma_f32_16x16x4_f32(false, a, false, b, (short)0, acc_ot, false, false);
        }
    }

    // --- extract diagonals ---------------------------------------------------
    // C/D layout: VGPR r -> M=r (lanes 0-15, N=lane) | M=r+8 (lanes 16-31, N=lane-16)
    // token i<8  : VGPR i,   lane i
    // token i>=8 : VGPR i-8, lane i+16
    const int rsel = (lane < 8) ? lane : ((lane >= 24) ? (lane - 24) : 0);
    float doo = acc_oo[0], dtt = acc_tt[0], dot = acc_ot[0];
    #pragma unroll
    for (int r = 1; r < 8; ++r) {
        doo = (rsel == r) ? acc_oo[r] : doo;
        dtt = (rsel == r) ? acc_tt[r] : dtt;
        dot = (rsel == r) ? acc_ot[r] : dot;
    }
    // gather token j's value into lane j (j = 0..15)
    const int src = (lane < 8) ? lane : (lane + 16);
    doo = __shfl(doo, src, 32);
    dtt = __shfl(dtt, src, 32);
    dot = __shfl(dot, src, 32);

    // --- per-token vMF scalar math (lanes 0..15) -----------------------------
    float nll = 0.f;
    if (lane < 16) {
        const int  tg    = targets[tokBase + lane];
        const float kappa = sqrtf(doo);
        const float tn    = sqrtf(dtt);
        const float cosv  = dot / (fmaxf(kappa, 1e-12f) * fmaxf(tn, 1e-12f));
        // v = m/2 - 1 = 255 ; s = sqrt((v+1)^2 + k^2) ; logcmk = s - (v-1)*log(v-1+s)
        const float s      = sqrtf(65536.0f + kappa * kappa);
        const float logcmk = s - 254.0f * logf(254.0f + s);
        nll = logcmk + log1pf(kappa) * (0.2f - cosv);
        if (tg == PAD_INDEX) nll = 0.f;
    }
    #pragma unroll
    for (int off = 16; off > 0; off >>= 1) nll += __shfl_xor(nll, off, 32);
    if (lane == 0) partial[tile] = nll;
}

// Fixed-order deterministic final reduction (no atomics).
__global__ __launch_bounds__(256) void vmf_reduce_kernel(
    const float* __restrict__ partial, float* __restrict__ out, int n)
{
    __shared__ float sm[256];
    float s = 0.f;
    for (int i = threadIdx.x; i < n; i += 256) s += partial[i];
    sm[threadIdx.x] = s;
    __syncthreads();
    #pragma unroll
    for (int off = 128; off > 0; off >>= 1) {
        if ((int)threadIdx.x < off) sm[threadIdx.x] += sm[threadIdx.x + off];
        __syncthreads();
    }
    if (threadIdx.x == 0) out[0] = sm[0];
}

extern "C" void kernel_launch(void* const* d_in, const int* in_sizes, int n_in,
                              void* d_out, int out_size, void* d_ws, size_t ws_size,
                              hipStream_t stream) {
    const float* outputs = (const float*)d_in[0];   // [32*512, 512] f32
    const int*   targets = (const int*)  d_in[1];   // [32*512] int32
    const float* table   = (const float*)d_in[2];   // [32000, 512] f32
    float*       out     = (float*)d_out;
    float*       partial = (float*)d_ws;

    const int ntok   = in_sizes[1];                 // 16384
    const int ntiles = ntok / 16;                   // 1024
    const int blocks = (ntiles + WAVES_PER_BLOCK - 1) / WAVES_PER_BLOCK;  // 128

    vmf_tile_kernel<<<dim3(blocks), dim3(256), 0, stream>>>(outputs, targets, table, partial, ntiles);
    vmf_reduce_kernel<<<dim3(1), dim3(256), 0, stream>>>(partial, out, ntiles);
}